// ThreeBodyDescriptor_35897336660167
// MI455X (gfx1250) — compile-verified
//
#include <hip/hip_runtime.h>

typedef float v2f __attribute__((ext_vector_type(2)));
typedef float v8f __attribute__((ext_vector_type(8)));

#define CUTOFF_F 5.0f

// hardware transcendental: v_log_f32 is log2, v_exp_f32 is exp2
#define LOG2F(x) __builtin_amdgcn_logf(x)
#define EXP2F(x) __builtin_amdgcn_exp2f(x)

// f = clamp(2*(1 - r/cutoff), min=0)
__device__ __forceinline__ float fcut(float r) {
    return fmaxf(0.0f, 2.0f - (2.0f / CUTOFF_F) * r);
}

// ---------------- K0: zero the per-atom counters ----------------
__global__ void zero_kernel(int* __restrict__ cursor, int N) {
    int i = blockIdx.x * blockDim.x + threadIdx.x;
    if (i < N) cursor[i] = 0;
}

// ---------------- K1: histogram of segment sizes ----------------
__global__ void count_kernel(const int* __restrict__ I, int* __restrict__ cnt, int E) {
    int e = blockIdx.x * blockDim.x + threadIdx.x;
    if (e < E) atomicAdd(&cnt[I[e]], 1);
}

// ---------------- K2: single-workgroup exclusive scan ----------------
// counts live in `cursor`; writes exclusive prefix to both off[] and cursor[]
__global__ void scan_kernel(int* __restrict__ cursor, int* __restrict__ off,
                            int N, int E) {
    __shared__ int sd[1024];
    __shared__ int s_run;
    const int tid = threadIdx.x;
    if (tid == 0) s_run = 0;
    __syncthreads();
    for (int base = 0; base < N; base += 1024) {
        int idx = base + tid;
        int x = (idx < N) ? cursor[idx] : 0;
        sd[tid] = x;
        __syncthreads();
        // Hillis-Steele inclusive scan
        for (int o = 1; o < 1024; o <<= 1) {
            int v = (tid >= o) ? sd[tid - o] : 0;
            __syncthreads();
            sd[tid] += v;
            __syncthreads();
        }
        int incl = sd[tid];
        int run  = s_run;
        if (idx < N) {
            int ex = run + incl - x;
            off[idx]    = ex;
            cursor[idx] = ex;   // becomes the scatter cursor
        }
        __syncthreads();
        if (tid == 1023) s_run = run + incl;
        __syncthreads();
    }
    if (tid == 0) off[N] = E;
}

// ---------------- K3: scatter per-edge log-features into segment order ----
// Stores t = {log2 f(r_jk), log2(f(r_ij)*f(r_ik))} per edge (float2).
// Species mask folded in: masked edges store t.y = -inf so exp2 -> 0.
__global__ void scatter_kernel(const int* __restrict__ I, const int* __restrict__ J,
                               const int* __restrict__ K,
                               const float* __restrict__ RIJ,
                               const float* __restrict__ RIK,
                               const float* __restrict__ RJK,
                               const int* __restrict__ Z,
                               int* __restrict__ cursor,
                               float2* __restrict__ t_s, int E) {
    int e = blockIdx.x * blockDim.x + threadIdx.x;
    if (e >= E) return;
    int n = I[e];
    bool m = (Z[n] == 1) && (Z[J[e]] == 1) && (Z[K[e]] == 1);
    int pos = atomicAdd(&cursor[n], 1);
    float tjk = LOG2F(fcut(RJK[e]));
    float tg  = m ? LOG2F(fcut(RIJ[e]) * fcut(RIK[e])) : -__builtin_inff();
    t_s[pos] = make_float2(tjk, tg);
}

// ---------------- K4: per-atom segment reduction via WMMA ----------------
// One wave32 per atom. D[a][b] += sum_k e_jk[k][a] * g[k][b], K=4 edges/step.
// A (16x4 f32): lanes 0-15 M=lane (K=0 in v0, K=1 in v1); lanes 16-31 K=2,3.
// B (4x16 f32): mirrored K pattern, N=lane&15. Rows/cols 8-15 fed zero.
// Fully branch-free body: clamped loads + select-to-zero, EXEC stays all-1s.
__global__ void segment_wmma_kernel(const int* __restrict__ off,
                                    const float2* __restrict__ t_s,
                                    float* __restrict__ out, int N, int E) {
    const int wave_in_blk = threadIdx.x >> 5;
    const int n = blockIdx.x * (blockDim.x >> 5) + wave_in_blk;  // atom id
    if (n >= N) return;                                          // wave-uniform

    const int lane = threadIdx.x & 31;
    const int half = lane >> 4;       // which pair of K slots this lane feeds
    const int fa   = lane & 15;       // feature row (A) / feature col (B)
    const bool fv  = (fa < 8);
    // exps[p] = 2*4^(p/7) = 2^(1 + 2p/7)
    const float efa = EXP2F(1.0f + (2.0f / 7.0f) * (float)fa);

    const int begin = off[n];
    const int end   = off[n + 1];

    v8f c = {0.f, 0.f, 0.f, 0.f, 0.f, 0.f, 0.f, 0.f};

    for (int kk = begin; kk < end; kk += 4) {
        const int e0 = kk + half * 2;
        const int e1 = e0 + 1;
        // clamped loads: never OOB, 16 lanes hit the same 8B -> one request
        float2 t0 = t_s[min(e0, E - 1)];
        float2 t1 = t_s[min(e1, E - 1)];
        const bool v0 = fv && (e0 < end);
        const bool v1 = fv && (e1 < end);
        float a0 = v0 ? EXP2F(efa * t0.x) : 0.0f;  // e_jk feature fa
        float b0 = v0 ? EXP2F(efa * t0.y) : 0.0f;  // (e_ij*e_ik) feature fa
        float a1 = v1 ? EXP2F(efa * t1.x) : 0.0f;
        float b1 = v1 ? EXP2F(efa * t1.y) : 0.0f;
        v2f A = {a0, a1};
        v2f B = {b0, b1};
        c = __builtin_amdgcn_wmma_f32_16x16x4_f32(
                /*neg_a=*/false, A, /*neg_b=*/false, B,
                /*c_mod=*/(short)0, c, /*reuse_a=*/false, /*reuse_b=*/false);
    }

    // D layout: lane l in 0..15 holds D[r][l] in c[r]. Valid block: r<8, l<8.
    if (lane < 8) {
        float* o = out + (size_t)n * 64;
#pragma unroll
        for (int r = 0; r < 8; ++r) o[r * 8 + lane] = c[r];  // out[n, r*8 + col]
    }
}

// ---------------- launcher ----------------
extern "C" void kernel_launch(void* const* d_in, const int* in_sizes, int n_in,
                              void* d_out, int out_size, void* d_ws, size_t ws_size,
                              hipStream_t stream) {
    const int* I   = (const int*)d_in[0];
    const int* J   = (const int*)d_in[1];
    const int* K   = (const int*)d_in[2];
    const float* RIJ = (const float*)d_in[3];
    const float* RIK = (const float*)d_in[4];
    const float* RJK = (const float*)d_in[5];
    const int* Z   = (const int*)d_in[6];
    float* out = (float*)d_out;

    const int E = in_sizes[0];
    const int N = in_sizes[6];

    // workspace layout (float2 region 8B-aligned: 2N+2 ints before it)
    int*    cursor = (int*)d_ws;                // N ints (counts, then cursor)
    int*    off    = cursor + N;                // N+1 ints (+1 pad for alignment)
    float2* t_s    = (float2*)(cursor + 2 * N + 2);  // E float2

    const int TB = 256;
    zero_kernel<<<(N + TB - 1) / TB, TB, 0, stream>>>(cursor, N);
    count_kernel<<<(E + TB - 1) / TB, TB, 0, stream>>>(I, cursor, E);
    scan_kernel<<<1, 1024, 0, stream>>>(cursor, off, N, E);
    scatter_kernel<<<(E + TB - 1) / TB, TB, 0, stream>>>(
        I, J, K, RIJ, RIK, RJK, Z, cursor, t_s, E);
    // 8 waves per block, one wave per atom
    segment_wmma_kernel<<<(N + 7) / 8, TB, 0, stream>>>(off, t_s, out, N, E);
}